// ResonanceField_6012954214414
// MI455X (gfx1250) — compile-verified
//
#include <hip/hip_runtime.h>

// ResonanceField: out[b,t,d] = x[b,t,d] * (1/H) * sum_h amp[h]*sin(2*pi*f[h]*(time + t/S))
// Pure bandwidth-bound row-scale. 256 MiB HBM traffic, ~11.5 us floor at 23.3 TB/s.

typedef __attribute__((ext_vector_type(4))) float v4f;

namespace {
constexpr int   kS     = 4096;              // sequence length
constexpr int   kD     = 1024;              // feature dim
constexpr int   kDV    = kD / 4;            // float4s per row == 256
constexpr int   kBlock = 256;               // blockDim == kDV -> one row per chunk per block
constexpr int   kChunk = 4;                 // rows (chunks) per block
constexpr float kTwoPi = 6.28318530717958647692f;
}

// ---------------- kernel 1: 16 KiB wave table (L2-resident) ----------------
__global__ __launch_bounds__(256) void wave_table_kernel(
    const float* __restrict__ freq, const float* __restrict__ amp,
    const float* __restrict__ time_p, float* __restrict__ wave, int H) {
  int t = blockIdx.x * blockDim.x + threadIdx.x;
  if (t >= kS) return;
  float tt = *time_p + (float)t * (1.0f / (float)kS);
  float acc = 0.0f;
  for (int h = 0; h < H; ++h) {
    // phases reach ~3.3e4 rad -> need full range reduction (sinf, not __sinf)
    acc += amp[h] * sinf(kTwoPi * freq[h] * tt);
  }
  wave[t] = acc / (float)H;
}

// ---------------- kernel 2: streaming NT b128 row-scale --------------------
// blockDim==256 and kDV==256 => for chunk j, the whole block reads one row:
//   row_j = blockIdx.x + j*gridDim.x   (uniform -> scalar s_load of wave)
__global__ __launch_bounds__(kBlock) void resonance_scale_kernel(
    const v4f* __restrict__ x, const float* __restrict__ wave,
    v4f* __restrict__ out, int nvec) {
  const int T   = (int)gridDim.x * kBlock;
  const int tid = (int)blockIdx.x * kBlock + (int)threadIdx.x;
  const int i0 = tid;
  const int i1 = tid + T;
  const int i2 = tid + 2 * T;
  const int i3 = tid + 3 * T;

  if (i3 < nvec) {
    // wave-uniform indices (function of blockIdx/gridDim only) -> SMEM loads
    const float w0 = wave[((int)blockIdx.x + 0 * (int)gridDim.x) & (kS - 1)];
    const float w1 = wave[((int)blockIdx.x + 1 * (int)gridDim.x) & (kS - 1)];
    const float w2 = wave[((int)blockIdx.x + 2 * (int)gridDim.x) & (kS - 1)];
    const float w3 = wave[((int)blockIdx.x + 3 * (int)gridDim.x) & (kS - 1)];

    // 4 independent NT streams in flight
    v4f a0 = __builtin_nontemporal_load(&x[i0]);
    v4f a1 = __builtin_nontemporal_load(&x[i1]);
    v4f a2 = __builtin_nontemporal_load(&x[i2]);
    v4f a3 = __builtin_nontemporal_load(&x[i3]);

    const v4f s0 = {w0, w0, w0, w0};
    const v4f s1 = {w1, w1, w1, w1};
    const v4f s2 = {w2, w2, w2, w2};
    const v4f s3 = {w3, w3, w3, w3};
    a0 *= s0; a1 *= s1; a2 *= s2; a3 *= s3;

    __builtin_nontemporal_store(a0, &out[i0]);
    __builtin_nontemporal_store(a1, &out[i1]);
    __builtin_nontemporal_store(a2, &out[i2]);
    __builtin_nontemporal_store(a3, &out[i3]);
  } else {
    // tail (only taken if sizes are not the reference's exact multiples)
    for (int i = i0; i < nvec; i += T) {
      const float w = wave[(i / kDV) & (kS - 1)];
      v4f a = __builtin_nontemporal_load(&x[i]);
      const v4f s = {w, w, w, w};
      a *= s;
      __builtin_nontemporal_store(a, &out[i]);
    }
  }
}

// -------- fallback: no workspace; recompute wave per block (still uniform) --
__device__ __forceinline__ float wave_val(const float* __restrict__ freq,
                                          const float* __restrict__ amp,
                                          float time, int t, int H) {
  float tt = time + (float)t * (1.0f / (float)kS);
  float acc = 0.0f;
  for (int h = 0; h < H; ++h) acc += amp[h] * sinf(kTwoPi * freq[h] * tt);
  return acc / (float)H;
}

__global__ __launch_bounds__(kBlock) void resonance_scale_inline_kernel(
    const v4f* __restrict__ x, const float* __restrict__ freq,
    const float* __restrict__ amp, const float* __restrict__ time_p,
    v4f* __restrict__ out, int nvec, int H) {
  const int T   = (int)gridDim.x * kBlock;
  const int tid = (int)blockIdx.x * kBlock + (int)threadIdx.x;
  const float time = *time_p;

  const int i0 = tid;
  const int i3 = tid + 3 * T;
  if (i3 < nvec) {
    const float w0 = wave_val(freq, amp, time, ((int)blockIdx.x + 0 * (int)gridDim.x) & (kS - 1), H);
    const float w1 = wave_val(freq, amp, time, ((int)blockIdx.x + 1 * (int)gridDim.x) & (kS - 1), H);
    const float w2 = wave_val(freq, amp, time, ((int)blockIdx.x + 2 * (int)gridDim.x) & (kS - 1), H);
    const float w3 = wave_val(freq, amp, time, ((int)blockIdx.x + 3 * (int)gridDim.x) & (kS - 1), H);
    v4f a0 = __builtin_nontemporal_load(&x[i0]);
    v4f a1 = __builtin_nontemporal_load(&x[i0 + T]);
    v4f a2 = __builtin_nontemporal_load(&x[i0 + 2 * T]);
    v4f a3 = __builtin_nontemporal_load(&x[i3]);
    const v4f s0 = {w0, w0, w0, w0};
    const v4f s1 = {w1, w1, w1, w1};
    const v4f s2 = {w2, w2, w2, w2};
    const v4f s3 = {w3, w3, w3, w3};
    a0 *= s0; a1 *= s1; a2 *= s2; a3 *= s3;
    __builtin_nontemporal_store(a0, &out[i0]);
    __builtin_nontemporal_store(a1, &out[i0 + T]);
    __builtin_nontemporal_store(a2, &out[i0 + 2 * T]);
    __builtin_nontemporal_store(a3, &out[i3]);
  } else {
    for (int i = i0; i < nvec; i += T) {
      const float w = wave_val(freq, amp, time, (i / kDV) & (kS - 1), H);
      v4f a = __builtin_nontemporal_load(&x[i]);
      const v4f s = {w, w, w, w};
      a *= s;
      __builtin_nontemporal_store(a, &out[i]);
    }
  }
}

extern "C" void kernel_launch(void* const* d_in, const int* in_sizes, int n_in,
                              void* d_out, int out_size, void* d_ws, size_t ws_size,
                              hipStream_t stream) {
  const float* x      = (const float*)d_in[0];
  const float* freq   = (const float*)d_in[1];
  const float* amp    = (const float*)d_in[2];
  const float* time_p = (const float*)d_in[3];
  float*       out    = (float*)d_out;

  const int H    = in_sizes[1];              // 12
  const int nvec = out_size / 4;             // float4 count = 8,388,608
  const int blocks = (nvec + kBlock * kChunk - 1) / (kBlock * kChunk);  // 8192

  if (ws_size >= (size_t)kS * sizeof(float)) {
    float* wave = (float*)d_ws;
    wave_table_kernel<<<(kS + 255) / 256, 256, 0, stream>>>(freq, amp, time_p, wave, H);
    resonance_scale_kernel<<<blocks, kBlock, 0, stream>>>(
        (const v4f*)x, wave, (v4f*)out, nvec);
  } else {
    resonance_scale_inline_kernel<<<blocks, kBlock, 0, stream>>>(
        (const v4f*)x, freq, amp, time_p, (v4f*)out, nvec, H);
  }
}